// GraphSAGE_79748952752809
// MI455X (gfx1250) — compile-verified
//
#include <hip/hip_runtime.h>

// GraphSAGE 2-layer forward for MI455X (gfx1250, wave32).
// f32 end-to-end; GEMMs via V_WMMA_F32_16X16X4_F32 (exact f32 matrix pipe).

typedef float v2f __attribute__((ext_vector_type(2)));
typedef float v8f __attribute__((ext_vector_type(8)));

#define IN_CH 602
#define HID   256
#define N1C   10240
#define N2C   1024

__device__ __forceinline__ v8f wmma_f32_k4(v2f a, v2f b, v8f c) {
  // 8 args: (neg_a, A, neg_b, B, c_mod, C, reuse_a, reuse_b)
  return __builtin_amdgcn_wmma_f32_16x16x4_f32(
      false, a, false, b, (short)0, c, false, false);
}

// ---------------------------------------------------------------------------
// Edge scatter: ssum[dst[e], :] += xsrc[src[e], :];  cnt[dst[e]] += 1
// One block per edge; atomics land in L2 (ssum fits easily in 192MB L2).
// ---------------------------------------------------------------------------
__global__ __launch_bounds__(256) void scatter_mean_accum(
    const float* __restrict__ xsrc, const int* __restrict__ src,
    const int* __restrict__ dst, float* __restrict__ ssum,
    float* __restrict__ cnt, int C) {
  const int e = blockIdx.x;
  const int s = src[e];
  const int d = dst[e];
  const float* xr = xsrc + (size_t)s * C;
  float* so = ssum + (size_t)d * C;
  for (int c = threadIdx.x; c < C; c += blockDim.x)
    atomicAdd(&so[c], xr[c]);
  if (threadIdx.x == 0) atomicAdd(&cnt[d], 1.0f);
}

// ---------------------------------------------------------------------------
// Fused SAGE layer:
//   out[m, n] = act( (ssum[m,:] / max(cnt[m],1)) @ Wl  + bl[n] + xroot[m,:] @ Wr )
// M-tile of 16 rows per block, 8 waves * 16 cols = 128 cols per block.
// K staged through LDS in chunks of 128 (row stride padded to 130).
// ---------------------------------------------------------------------------
template<int K, bool RELU>
__global__ __launch_bounds__(256) void sage_fused_gemm(
    const float* __restrict__ ssum, const float* __restrict__ cnt,
    const float* __restrict__ xroot,
    const float* __restrict__ Wl, const float* __restrict__ bl,
    const float* __restrict__ Wr, float* __restrict__ out) {
  constexpr int KC  = 128;      // K chunk
  constexpr int KCP = 130;      // padded LDS row stride (bank-conflict avoidance)
  __shared__ float ldsA[16 * KCP];   // scaled neighbor aggregate
  __shared__ float ldsX[16 * KCP];   // root features
  __shared__ float scale16[16];

  const int m0 = blockIdx.x * 16;
  if (threadIdx.x < 16) {
    float cv = cnt[m0 + threadIdx.x];
    scale16[threadIdx.x] = 1.0f / fmaxf(cv, 1.0f);
  }

  const int lane  = threadIdx.x & 31;
  const int wave  = threadIdx.x >> 5;
  const int row16 = lane & 15;
  const int koff  = (lane >> 4) << 1;                  // 0 or 2
  const int col   = blockIdx.y * 128 + wave * 16 + row16;

  v8f acc = {};

  for (int kbase = 0; kbase < K; kbase += KC) {
    const int klen = (K - kbase < KC) ? (K - kbase) : KC;   // uniform over block

    __syncthreads();   // previous chunk fully consumed (also covers scale16 init)
    for (int idx = threadIdx.x; idx < 16 * klen; idx += 256) {
      int r = idx / klen;
      int c = idx - r * klen;
      size_t g = (size_t)(m0 + r) * K + kbase + c;
      ldsA[r * KCP + c] = ssum[g] * scale16[r];
      ldsX[r * KCP + c] = xroot[g];
    }
    __syncthreads();

    const int kq = klen & ~3;   // full K=4 quads in this chunk
    if (klen == KC) {
      #pragma unroll 8
      for (int k0 = 0; k0 < KC; k0 += 4) {
        const int kr = kbase + k0 + koff;
        v2f a  = *(const v2f*)&ldsA[row16 * KCP + k0 + koff];
        v2f ax = *(const v2f*)&ldsX[row16 * KCP + k0 + koff];
        v2f b, bx;
        b.x  = Wl[(size_t)kr * HID + col];
        b.y  = Wl[(size_t)(kr + 1) * HID + col];
        bx.x = Wr[(size_t)kr * HID + col];
        bx.y = Wr[(size_t)(kr + 1) * HID + col];
        acc = wmma_f32_k4(a, b, acc);
        acc = wmma_f32_k4(ax, bx, acc);
      }
    } else {
      for (int k0 = 0; k0 < kq; k0 += 4) {
        const int kr = kbase + k0 + koff;
        v2f a  = *(const v2f*)&ldsA[row16 * KCP + k0 + koff];
        v2f ax = *(const v2f*)&ldsX[row16 * KCP + k0 + koff];
        v2f b, bx;
        b.x  = Wl[(size_t)kr * HID + col];
        b.y  = Wl[(size_t)(kr + 1) * HID + col];
        bx.x = Wr[(size_t)kr * HID + col];
        bx.y = Wr[(size_t)(kr + 1) * HID + col];
        acc = wmma_f32_k4(a, b, acc);
        acc = wmma_f32_k4(ax, bx, acc);
      }
      if (klen & 3) {   // K % 4 == 2 (e.g. K=602): zero-pad K on lanes 16..31
        v2f a = {0.f, 0.f}, b = {0.f, 0.f}, ax = {0.f, 0.f}, bx = {0.f, 0.f};
        if (koff == 0) {
          const int kr = kbase + kq;
          a  = *(const v2f*)&ldsA[row16 * KCP + kq];
          ax = *(const v2f*)&ldsX[row16 * KCP + kq];
          b.x  = Wl[(size_t)kr * HID + col];
          b.y  = Wl[(size_t)(kr + 1) * HID + col];
          bx.x = Wr[(size_t)kr * HID + col];
          bx.y = Wr[(size_t)(kr + 1) * HID + col];
        }
        acc = wmma_f32_k4(a, b, acc);     // EXEC all-ones; padded lanes add 0
        acc = wmma_f32_k4(ax, bx, acc);
      }
    }
  }

  // Epilogue: C/D layout — VGPR v: row = v + 8*(lane>=16), col = lane&15.
  const float bias = bl[col];
  const int rbase = m0 + ((lane >> 4) << 3);
  #pragma unroll
  for (int v = 0; v < 8; v++) {
    float rv = acc[v] + bias;
    if (RELU) rv = fmaxf(rv, 0.0f);
    out[(size_t)(rbase + v) * HID + col] = rv;
  }
}

// ---------------------------------------------------------------------------
// Launch. Workspace layout:
//   [ ssum1 (10240*602) | cnt1 (10240) | ssum2 (1024*256) | cnt2 (1024) | h (10240*256) ]
// ---------------------------------------------------------------------------
extern "C" void kernel_launch(void* const* d_in, const int* in_sizes, int n_in,
                              void* d_out, int out_size, void* d_ws, size_t ws_size,
                              hipStream_t stream) {
  const float* x   = (const float*)d_in[0];
  const int* src1  = (const int*)d_in[1];
  const int* dst1  = (const int*)d_in[2];
  const int* src2  = (const int*)d_in[3];
  const int* dst2  = (const int*)d_in[4];
  const float* W1l = (const float*)d_in[5];
  const float* b1  = (const float*)d_in[6];
  const float* W1r = (const float*)d_in[7];
  const float* W2l = (const float*)d_in[8];
  const float* b2  = (const float*)d_in[9];
  const float* W2r = (const float*)d_in[10];
  const int E1 = in_sizes[1];
  const int E2 = in_sizes[3];

  float* ssum1 = (float*)d_ws;
  float* cnt1  = ssum1 + (size_t)N1C * IN_CH;
  float* ssum2 = cnt1 + N1C;
  float* cnt2  = ssum2 + (size_t)N2C * HID;
  float* h     = cnt2 + N2C;

  const size_t zero_bytes =
      ((size_t)N1C * IN_CH + N1C + (size_t)N2C * HID + N2C) * sizeof(float);
  hipMemsetAsync(d_ws, 0, zero_bytes, stream);

  // Layer 1
  scatter_mean_accum<<<E1, 256, 0, stream>>>(x, src1, dst1, ssum1, cnt1, IN_CH);
  sage_fused_gemm<IN_CH, true><<<dim3(N1C / 16, HID / 128), 256, 0, stream>>>(
      ssum1, cnt1, x, W1l, b1, W1r, h);

  // Layer 2
  scatter_mean_accum<<<E2, 256, 0, stream>>>(h, src2, dst2, ssum2, cnt2, HID);
  sage_fused_gemm<HID, false><<<dim3(N2C / 16, HID / 128), 256, 0, stream>>>(
      ssum2, cnt2, h, W2l, b2, W2r, (float*)d_out);
}